// STConvClassifier_38293928411455
// MI455X (gfx1250) — compile-verified
//
#include <hip/hip_runtime.h>
#include <stdint.h>

// ---------------- problem constants (match reference) ----------------
#define B_   2
#define T_   12
#define N_   50000
#define H_   16
#define KT_  3
#define E_   800000
#define T1_  10          // after tconv1
#define T2_  8           // after tconv2
#define NC_  2
#define NTILE_ (N_ / 16) // 3125, exact

typedef float v2f __attribute__((ext_vector_type(2)));
typedef float v8f __attribute__((ext_vector_type(8)));

// D = A(16x4 f32) * B(4x16 f32) + C(16x16 f32)
#define WMMA_F32(a, b, c) \
    __builtin_amdgcn_wmma_f32_16x16x4_f32(false, (a), false, (b), (short)0, (c), false, false)

// ---------------- kernel 0: zero atomic-target scratch ----------------
__global__ void k_zero(float* __restrict__ p, int n) {
    int i = blockIdx.x * blockDim.x + threadIdx.x;
    if (i < n) p[i] = 0.0f;
}

// ---------------- kernel 1: temporal conv 1 (C_IN=1) ----------------
// h1[((b*T1+t)*N+n)*16+h] = relu(P * sigmoid(Q) + R)
__global__ void k_tconv1(const float* __restrict__ x,
                         const float* __restrict__ wP, const float* __restrict__ bP,
                         const float* __restrict__ wQ, const float* __restrict__ bQ,
                         const float* __restrict__ wR, const float* __restrict__ bR,
                         float* __restrict__ h1) {
    int idx = blockIdx.x * blockDim.x + threadIdx.x;   // B*T1*N*H = 16M
    if (idx >= B_ * T1_ * N_ * H_) return;
    int h   = idx & 15;
    int btn = idx >> 4;
    int n   = btn % N_;
    int bt  = btn / N_;
    int t   = bt % T1_;
    int b   = bt / T1_;
    float p = bP[h], q = bQ[h], r = bR[h];
#pragma unroll
    for (int k = 0; k < KT_; ++k) {
        float xv = x[(b * T_ + t + k) * N_ + n];       // C_IN = 1
        p += xv * wP[k * 16 + h];                      // w: (3,1,16)
        q += xv * wQ[k * 16 + h];
        r += xv * wR[k * 16 + h];
    }
    float s = 1.0f / (1.0f + __expf(-q));
    float v = p * s + r;
    h1[idx] = v > 0.0f ? v : 0.0f;
}

// ---------------- kernel 2: degree = segment_sum(w, src) ----------------
__global__ void k_deg(const long long* __restrict__ ei, const float* __restrict__ ew,
                      float* __restrict__ deg) {
    int e = blockIdx.x * blockDim.x + threadIdx.x;
    if (e >= E_) return;
    atomicAdd(&deg[(int)ei[e]], ew[e]);
}

// ---------------- kernel 3: per-edge normalized weight ----------------
__global__ void k_norm(const long long* __restrict__ ei, const float* __restrict__ ew,
                       const float* __restrict__ deg, float* __restrict__ norm) {
    int e = blockIdx.x * blockDim.x + threadIdx.x;
    if (e >= E_) return;
    int s = (int)ei[e];
    int d = (int)ei[E_ + e];
    float ds = deg[s], dd = deg[d];
    float is = ds > 0.0f ? rsqrtf(fmaxf(ds, 1e-12f)) : 0.0f;
    float id = dd > 0.0f ? rsqrtf(fmaxf(dd, 1e-12f)) : 0.0f;
    norm[e] = -is * ew[e] * id;
}

// ---------------- kernel 4: SpMM scatter agg[dst,f] += norm*h1[src,f] ----------------
// agg is (N, B*T1*H=320); f = bt*16 + h; thread = (edge, 4-feature group)
__global__ void k_spmm(const long long* __restrict__ ei, const float* __restrict__ norm,
                       const float* __restrict__ h1, float* __restrict__ agg) {
    long long tid = (long long)blockIdx.x * blockDim.x + threadIdx.x;
    if (tid >= (long long)E_ * 80) return;
    int e = (int)(tid / 80);
    int g = (int)(tid % 80);
    int f  = g * 4;
    int bt = f >> 4;
    int h  = f & 15;
    int s = (int)ei[e];
    int d = (int)ei[E_ + e];
    float w = norm[e];
    const float4 xv = *(const float4*)&h1[(bt * N_ + s) * 16 + h];
    float* o = &agg[(long long)d * 320 + f];
    atomicAdd(o + 0, w * xv.x);
    atomicAdd(o + 1, w * xv.y);
    atomicAdd(o + 2, w * xv.z);
    atomicAdd(o + 3, w * xv.w);
}

// ---------------- kernel 5: ChebConv dense part via f32 WMMA ----------------
// hc = relu(h1 @ w0 + tx1 @ w1 + b); one wave = one (b*t, 16-node tile)
__global__ void k_cheb(const float* __restrict__ h1, const float* __restrict__ agg,
                       const float* __restrict__ cw, const float* __restrict__ cb,
                       float* __restrict__ hc) {
    int lane = threadIdx.x & 31;
    int wave = (blockIdx.x * blockDim.x + threadIdx.x) >> 5;  // 0..62499 exact
    int tile = wave % NTILE_;
    int bt   = wave / NTILE_;                                  // 0..19 (= b*10+t)
    int m  = lane & 15;
    int hf = lane >> 4;

    v8f acc = {};
    const float* xb = h1 + ((long long)bt * N_ + tile * 16) * 16;  // X[node][c]
    const float* ab = agg + (long long)(tile * 16) * 320 + bt * 16; // tx1: stride 320/node
#pragma unroll
    for (int kk = 0; kk < 4; ++kk) {           // Tx_0 @ W0
        int c0 = kk * 4 + hf * 2;
        v2f a;  a.x = xb[m * 16 + c0];          a.y = xb[m * 16 + c0 + 1];
        v2f bb; bb.x = cw[c0 * 16 + m];         bb.y = cw[(c0 + 1) * 16 + m];
        acc = WMMA_F32(a, bb, acc);
    }
#pragma unroll
    for (int kk = 0; kk < 4; ++kk) {           // Tx_1 @ W1
        int c0 = kk * 4 + hf * 2;
        v2f a;  a.x = ab[m * 320 + c0];         a.y = ab[m * 320 + c0 + 1];
        v2f bb; bb.x = cw[256 + c0 * 16 + m];   bb.y = cw[256 + (c0 + 1) * 16 + m];
        acc = WMMA_F32(a, bb, acc);
    }
    float bias = cb[m];
    float* out = hc + ((long long)bt * N_ + tile * 16) * 16;
#pragma unroll
    for (int v = 0; v < 8; ++v) {
        float val = acc[v] + bias;
        out[(v + hf * 8) * 16 + m] = val > 0.0f ? val : 0.0f;
    }
}

// ---------------- kernel 6: temporal conv 2 + mean + linear, WMMA ----------------
// one wave = one (b, 16-node tile); K = KT*H = 48 -> 12 WMMAs/branch/timestep
__global__ void k_tconv2(const float* __restrict__ hc,
                         const float* __restrict__ wP, const float* __restrict__ bP,
                         const float* __restrict__ wQ, const float* __restrict__ bQ,
                         const float* __restrict__ wR, const float* __restrict__ bR,
                         const float* __restrict__ linw, const float* __restrict__ linb,
                         float* __restrict__ out) {
    int lane = threadIdx.x & 31;
    int wave = (blockIdx.x * blockDim.x + threadIdx.x) >> 5;
    if (wave >= B_ * NTILE_) return;           // uniform per-wave exit (EXEC stays full)
    int tile = wave % NTILE_;
    int b    = wave / NTILE_;
    int m  = lane & 15;
    int hf = lane >> 4;

    // Preload all weights in B-matrix layout: 12 K-chunks x v2f per branch.
    v2f Wp[12], Wq[12], Wr[12];
#pragma unroll
    for (int kk = 0; kk < 12; ++kk) {
        int K0 = kk * 4 + hf * 2;                          // K in [0,48)
        int i0 = ((K0 >> 4) * 16 + (K0 & 15)) * 16 + m;    // (k,c,h) of (3,16,16)
        int i1 = (((K0 + 1) >> 4) * 16 + ((K0 + 1) & 15)) * 16 + m;
        Wp[kk].x = wP[i0]; Wp[kk].y = wP[i1];
        Wq[kk].x = wQ[i0]; Wq[kk].y = wQ[i1];
        Wr[kk].x = wR[i0]; Wr[kk].y = wR[i1];
    }
    float bp = bP[m], bq = bQ[m], br = bR[m];

    v8f sum = {};
    for (int t = 0; t < T2_; ++t) {
        v8f ap, aq, ar;
#pragma unroll
        for (int v = 0; v < 8; ++v) { ap[v] = bp; aq[v] = bq; ar[v] = br; }
#pragma unroll
        for (int kk = 0; kk < 12; ++kk) {
            int K0 = kk * 4 + hf * 2;
            int k = K0 >> 4;                   // tap (uniform within a 4-wide chunk)
            int c = K0 & 15;
            const float* xb = hc + (((long long)(b * T1_ + t + k)) * N_ + tile * 16 + m) * 16;
            v2f a; a.x = xb[c]; a.y = xb[c + 1];
            ap = WMMA_F32(a, Wp[kk], ap);
            aq = WMMA_F32(a, Wq[kk], aq);
            ar = WMMA_F32(a, Wr[kk], ar);
        }
#pragma unroll
        for (int v = 0; v < 8; ++v) {          // GLU gate + relu, accumulate for mean
            float s   = 1.0f / (1.0f + __expf(-aq[v]));
            float val = ap[v] * s + ar[v];
            sum[v] += val > 0.0f ? val : 0.0f;
        }
    }

    // mean over T2, linear 16->2 via intra-16-lane butterfly reduction
    float lw0 = linw[m * 2 + 0], lw1 = linw[m * 2 + 1];
    float lb0 = linb[0], lb1 = linb[1];
#pragma unroll
    for (int v = 0; v < 8; ++v) {
        float mean = sum[v] * (1.0f / (float)T2_);
        float c0 = mean * lw0;
        float c1 = mean * lw1;
#pragma unroll
        for (int off = 1; off < 16; off <<= 1) {
            c0 += __shfl_xor(c0, off, 32);
            c1 += __shfl_xor(c1, off, 32);
        }
        if (m == 0) {                           // lanes 0 and 16 write their half's rows
            int n = tile * 16 + v + hf * 8;
            out[(b * N_ + n) * 2 + 0] = c0 + lb0;
            out[(b * N_ + n) * 2 + 1] = c1 + lb1;
        }
    }
}

// ---------------- host-side orchestration ----------------
extern "C" void kernel_launch(void* const* d_in, const int* in_sizes, int n_in,
                              void* d_out, int out_size, void* d_ws, size_t ws_size,
                              hipStream_t stream) {
    const float*     x    = (const float*)d_in[0];
    const long long* ei   = (const long long*)d_in[1];   // int64 (2,E)
    const float*     ew   = (const float*)d_in[2];
    const float* tc1Pw = (const float*)d_in[3],  *tc1Pb = (const float*)d_in[4];
    const float* tc1Qw = (const float*)d_in[5],  *tc1Qb = (const float*)d_in[6];
    const float* tc1Rw = (const float*)d_in[7],  *tc1Rb = (const float*)d_in[8];
    const float* tc2Pw = (const float*)d_in[9],  *tc2Pb = (const float*)d_in[10];
    const float* tc2Qw = (const float*)d_in[11], *tc2Qb = (const float*)d_in[12];
    const float* tc2Rw = (const float*)d_in[13], *tc2Rb = (const float*)d_in[14];
    const float* chebw = (const float*)d_in[15], *chebb = (const float*)d_in[16];
    const float* linw  = (const float*)d_in[17], *linb  = (const float*)d_in[18];
    float* out = (float*)d_out;

    // workspace layout (floats): h1 | agg | deg | hc | norm   (~195 MB)
    float* ws   = (float*)d_ws;
    float* h1   = ws;                                       // 16,000,000
    float* agg  = h1  + (size_t)B_ * T1_ * N_ * H_;         // 16,000,000
    float* deg  = agg + (size_t)N_ * B_ * T1_ * H_;         //     50,000 (adjacent to agg)
    float* hc   = deg + N_;                                 // 16,000,000
    float* norm = hc  + (size_t)B_ * T1_ * N_ * H_;         //    800,000

    const int ZN = N_ * B_ * T1_ * H_ + N_;                 // agg+deg contiguous zero
    k_zero<<<(ZN + 255) / 256, 256, 0, stream>>>(agg, ZN);

    const int n1 = B_ * T1_ * N_ * H_;                      // 16M
    k_tconv1<<<(n1 + 255) / 256, 256, 0, stream>>>(x, tc1Pw, tc1Pb, tc1Qw, tc1Qb,
                                                   tc1Rw, tc1Rb, h1);

    k_deg<<<(E_ + 255) / 256, 256, 0, stream>>>(ei, ew, deg);
    k_norm<<<(E_ + 255) / 256, 256, 0, stream>>>(ei, ew, deg, norm);

    const long long nsp = (long long)E_ * 80;               // 64M threads
    k_spmm<<<(unsigned)((nsp + 255) / 256), 256, 0, stream>>>(ei, norm, h1, agg);

    const int cheb_waves = B_ * T1_ * NTILE_;               // 62500 -> 15625 blocks x 4 waves
    k_cheb<<<cheb_waves / 4, 128, 0, stream>>>(h1, agg, chebw, chebb, hc);

    const int tc2_waves = B_ * NTILE_;                      // 6250 waves
    k_tconv2<<<(tc2_waves * 32 + 127) / 128, 128, 0, stream>>>(hc, tc2Pw, tc2Pb, tc2Qw,
                                                               tc2Qb, tc2Rw, tc2Rb,
                                                               linw, linb, out);
    (void)in_sizes; (void)n_in; (void)out_size; (void)ws_size;
}